// LaplacianLoss_6528350290149
// MI455X (gfx1250) — compile-verified
//
#include <hip/hip_runtime.h>
#include <math.h>

typedef __attribute__((ext_vector_type(2))) float v2f;
typedef __attribute__((ext_vector_type(8))) float v8f;

#define EPSF 1e-12f
#define RED_BLOCKS 1024

__device__ __forceinline__ void atom_add_nr(float* p, float v) {
  // relaxed, agent scope, result discarded -> no-return global_atomic_add_f32
  __hip_atomic_fetch_add(p, v, __ATOMIC_RELAXED, __HIP_MEMORY_SCOPE_AGENT);
}

// Full wave32 sum computed on the matrix pipe.
// A (16x4 f32, 2 VGPRs): VGPR0 = lane value, VGPR1 = 0. B (4x16) = all ones.
// D[m][n] = sum_k A[m][k] = v[m] + v[m+16]  (K-slot placement irrelevant).
// Sum of D VGPRs 0..7 per lane = half-row total; shfl_xor(16) completes it.
__device__ __forceinline__ float wave_sum_wmma(float v) {
  v2f a; a[0] = v;    a[1] = 0.0f;
  v2f b; b[0] = 1.0f; b[1] = 1.0f;
  v8f c = {};
  v8f d = __builtin_amdgcn_wmma_f32_16x16x4_f32(
      /*neg_a=*/false, a, /*neg_b=*/false, b,
      /*c_mod=*/(short)0, c, /*reuse_a=*/false, /*reuse_b=*/false);
  float s = ((d[0] + d[1]) + (d[2] + d[3])) + ((d[4] + d[5]) + (d[6] + d[7]));
  s += __shfl_xor(s, 16, 32);
  return s;
}

__global__ void lap_zero_ws(float* __restrict__ p, long long n) {
  long long i = (long long)blockIdx.x * blockDim.x + threadIdx.x;
  long long stride = (long long)gridDim.x * blockDim.x;
  for (; i < n; i += stride) p[i] = 0.0f;
}

// Per face (a,b,c): accum[a] += (2, v[b]+v[c]); accum[b] += (2, v[c]+v[a]);
// accum[c] += (2, v[a]+v[b]).  Exactly the 6 directed edges of the reference,
// fused to 12 atomics/face instead of 24.
__global__ void lap_scatter(const float* __restrict__ verts,
                            const int* __restrict__ faces,
                            float* __restrict__ accum, int F) {
  int i = blockIdx.x * blockDim.x + threadIdx.x;
  if (i >= F) return;

  // faces are read exactly once: stream with NT + prefetch ahead
  __builtin_prefetch(faces + 3 * (i + 4096), 0, 0);
  int a = __builtin_nontemporal_load(faces + 3 * i + 0);
  int b = __builtin_nontemporal_load(faces + 3 * i + 1);
  int c = __builtin_nontemporal_load(faces + 3 * i + 2);

  // verts (24MB) are reused (avg degree ~12) and L2-resident: normal loads
  float ax = verts[3 * a + 0], ay = verts[3 * a + 1], az = verts[3 * a + 2];
  float bx = verts[3 * b + 0], by = verts[3 * b + 1], bz = verts[3 * b + 2];
  float cx = verts[3 * c + 0], cy = verts[3 * c + 1], cz = verts[3 * c + 2];

  float* pa = accum + 4 * a;
  atom_add_nr(pa + 0, 2.0f);
  atom_add_nr(pa + 1, bx + cx);
  atom_add_nr(pa + 2, by + cy);
  atom_add_nr(pa + 3, bz + cz);

  float* pb = accum + 4 * b;
  atom_add_nr(pb + 0, 2.0f);
  atom_add_nr(pb + 1, cx + ax);
  atom_add_nr(pb + 2, cy + ay);
  atom_add_nr(pb + 3, cz + az);

  float* pc = accum + 4 * c;
  atom_add_nr(pc + 0, 2.0f);
  atom_add_nr(pc + 1, ax + bx);
  atom_add_nr(pc + 2, ay + by);
  atom_add_nr(pc + 3, az + bz);
}

__global__ void lap_vertex_loss(const float* __restrict__ verts,
                                const float4* __restrict__ accum,
                                float* __restrict__ partials, int V) {
  float acc = 0.0f;
  int stride = gridDim.x * blockDim.x;
  for (int i = blockIdx.x * blockDim.x + threadIdx.x; i < V; i += stride) {
    float4 s = accum[i];            // (deg, sx, sy, sz) in one 16B load
    float deg = s.x;
    float has = (deg > 0.0f) ? 1.0f : 0.0f;
    float inv = has / fmaxf(deg, 1.0f);
    float vx = verts[3 * i + 0], vy = verts[3 * i + 1], vz = verts[3 * i + 2];
    float lx = fmaf(s.y, inv, -vx * has);
    float ly = fmaf(s.z, inv, -vy * has);
    float lz = fmaf(s.w, inv, -vz * has);
    acc += sqrtf(fmaf(lx, lx, fmaf(ly, ly, fmaf(lz, lz, EPSF))));
  }
  // no divergence here: every thread reaches the WMMA with EXEC all-1s
  float ws = wave_sum_wmma(acc);
  __shared__ float lds[8];
  int wid = threadIdx.x >> 5;
  if ((threadIdx.x & 31) == 0) lds[wid] = ws;
  __syncthreads();
  if (threadIdx.x == 0) {
    float t = 0.0f;
    int nw = blockDim.x >> 5;
    for (int w = 0; w < nw; ++w) t += lds[w];
    partials[blockIdx.x] = t;
  }
}

__global__ void lap_final(const float* __restrict__ partials, int np,
                          float* __restrict__ out, float inv_v) {
  float acc = 0.0f;
  for (int i = threadIdx.x; i < np; i += blockDim.x) acc += partials[i];
  float ws = wave_sum_wmma(acc);
  __shared__ float lds[8];
  int wid = threadIdx.x >> 5;
  if ((threadIdx.x & 31) == 0) lds[wid] = ws;
  __syncthreads();
  if (threadIdx.x == 0) {
    float t = 0.0f;
    int nw = blockDim.x >> 5;
    for (int w = 0; w < nw; ++w) t += lds[w];
    out[0] = t * inv_v;             // LOSS_WEIGHT == 1
  }
}

extern "C" void kernel_launch(void* const* d_in, const int* in_sizes, int n_in,
                              void* d_out, int out_size, void* d_ws, size_t ws_size,
                              hipStream_t stream) {
  const float* verts = (const float*)d_in[0];
  const int*   faces = (const int*)d_in[1];
  int V = in_sizes[0] / 3;
  int F = in_sizes[1] / 3;

  float* accum    = (float*)d_ws;                 // [V][4] = (deg, sx, sy, sz)
  float* partials = accum + (size_t)4 * V;        // [RED_BLOCKS]
  float* out      = (float*)d_out;

  lap_zero_ws<<<2048, 256, 0, stream>>>(accum, (long long)4 * V);
  lap_scatter<<<(F + 255) / 256, 256, 0, stream>>>(verts, faces, accum, F);
  lap_vertex_loss<<<RED_BLOCKS, 256, 0, stream>>>(verts, (const float4*)accum,
                                                  partials, V);
  lap_final<<<1, 256, 0, stream>>>(partials, RED_BLOCKS, out,
                                   1.0f / (float)V);
}